// LDHC_69853348102408
// MI455X (gfx1250) — compile-verified
//
#include <hip/hip_runtime.h>
#include <stdint.h>

// ---------------------------------------------------------------------------
// LightGCN propagation for MI455X (gfx1250, wave32).
//   d_out is [N, 256] row-major fp32; column block k (64 floats) = layer k.
//   Layer k SpMM reads col block k-1 and writes col block k (atomic-free CSR
//   path when workspace allows; float-atomic fallback otherwise).
// ---------------------------------------------------------------------------

#define EMB   64
#define OUTC  256          // 4 * EMB floats per output row
#define SCAN_T 256
#define SCAN_I 4
#define SCAN_TILE (SCAN_T * SCAN_I)   // 1024 elements per scan block

// Probe gfx1250 async global->LDS builtins (fallback compiles cleanly).
#if defined(__has_builtin)
#if __has_builtin(__builtin_amdgcn_global_load_async_to_lds_b64) && \
    __has_builtin(__builtin_amdgcn_s_wait_asynccnt)
#define USE_ASYNC_LDS 1
#endif
#endif

typedef float fvec4 __attribute__((ext_vector_type(4)));
typedef int   v2i   __attribute__((vector_size(8)));
typedef __attribute__((address_space(1))) v2i* g_v2i_p;   // global v2i*
typedef __attribute__((address_space(3))) v2i* l_v2i_p;   // LDS v2i*

// ----------------------------- CSR SpMM (atomic-free) -----------------------
// Hot kernel, placed first so the disasm snippet shows it.
// 16 lanes per destination row; float4 accumulator per lane (64 dims total).
// Row's packed (src,val) list staged into LDS via async global->LDS DMA
// (tracked by ASYNCcnt), then each edge is a 16-lane x float4 = 256B coalesced
// gather out of L2 feeding v_fma chains; one coalesced 256B store per row.
__global__ void lgcn_spmm_csr(float* out, const long long* __restrict__ pairs,
                              const int* __restrict__ rowptr, int n,
                              int colPrev, int colCur) {
#ifdef USE_ASYNC_LDS
  __shared__ long long stage[16 * 64];               // 8 KB: 64 edges x 16 groups
  int group = threadIdx.x >> 4;
#endif
  int g = blockIdx.x * blockDim.x + threadIdx.x;
  int row = g >> 4, lane = g & 15;
  if (row >= n) return;
  int beg = rowptr[row], end = rowptr[row + 1];
  const float* xb = out + colPrev + 4 * lane;        // + src*OUTC
  fvec4 acc = {0.f, 0.f, 0.f, 0.f};

#ifdef USE_ASYNC_LDS
  int deg = end - beg;
  int nst = deg < 64 ? deg : 64;
  for (int j = lane; j < nst; j += 16) {
    __builtin_amdgcn_global_load_async_to_lds_b64(
        (g_v2i_p)(uintptr_t)(pairs + beg + j),
        (l_v2i_p)(uint32_t)(uintptr_t)(&stage[(size_t)group * 64 + j]),
        0, 0);
  }
  __builtin_amdgcn_s_wait_asynccnt(0);
#pragma unroll 2
  for (int e = 0; e < nst; ++e) {
    long long c = stage[group * 64 + e];
    int s = (int)(unsigned)c;
    float v = __uint_as_float((unsigned)((unsigned long long)c >> 32));
    fvec4 gv = *(const fvec4*)(xb + (size_t)s * OUTC);
    acc.x = fmaf(v, gv.x, acc.x);
    acc.y = fmaf(v, gv.y, acc.y);
    acc.z = fmaf(v, gv.z, acc.z);
    acc.w = fmaf(v, gv.w, acc.w);
  }
  for (int e = beg + nst; e < end; ++e) {            // rare tail (deg > 64)
    long long c = pairs[e];
    int s = (int)(unsigned)c;
    float v = __uint_as_float((unsigned)((unsigned long long)c >> 32));
    fvec4 gv = *(const fvec4*)(xb + (size_t)s * OUTC);
    acc.x = fmaf(v, gv.x, acc.x);
    acc.y = fmaf(v, gv.y, acc.y);
    acc.z = fmaf(v, gv.z, acc.z);
    acc.w = fmaf(v, gv.w, acc.w);
  }
#else
  for (int e = beg; e < end; ++e) {
    __builtin_prefetch(pairs + e + 8, 0, 0);
    long long c = pairs[e];
    int s = (int)(unsigned)c;
    float v = __uint_as_float((unsigned)((unsigned long long)c >> 32));
    fvec4 gv = *(const fvec4*)(xb + (size_t)s * OUTC);
    acc.x = fmaf(v, gv.x, acc.x);
    acc.y = fmaf(v, gv.y, acc.y);
    acc.z = fmaf(v, gv.z, acc.z);
    acc.w = fmaf(v, gv.w, acc.w);
  }
#endif
  *(fvec4*)(out + (size_t)row * OUTC + colCur + 4 * lane) = acc;
}

// ----------------------------- init ----------------------------------------
// Writes col 0 = [user;item] embeddings, zeros cols 1..3 (needed only for the
// atomic fallback, but cheap streaming stores).
__global__ void lgcn_init(const float* __restrict__ ue, const float* __restrict__ ie,
                          float* __restrict__ out, int userN, int n) {
  int g = blockIdx.x * blockDim.x + threadIdx.x;
  int node = g >> 4, lane = g & 15;
  if (node >= n) return;
  const fvec4* s = (node < userN)
      ? (const fvec4*)(ue + (size_t)node * EMB)
      : (const fvec4*)(ie + (size_t)(node - userN) * EMB);
  fvec4 v = __builtin_nontemporal_load(s + lane);   // one-shot stream: NT
  fvec4 z = {0.f, 0.f, 0.f, 0.f};
  fvec4* o = (fvec4*)(out + (size_t)node * OUTC) + lane;
  o[0]  = v;   // col 0
  o[16] = z;   // col 1
  o[32] = z;   // col 2
  o[48] = z;   // col 3
}

// ----------------------------- CSR build -----------------------------------
__global__ void lgcn_hist(const int* __restrict__ dst, int* __restrict__ cnt, int nnz) {
  int i = blockIdx.x * blockDim.x + threadIdx.x;
  if (i < nnz) atomicAdd(&cnt[dst[i]], 1);
}

__device__ __forceinline__ int block_excl_scan_256(int tsum, int* sm) {
  int tid = threadIdx.x;
  sm[tid] = tsum;
  __syncthreads();
  for (int off = 1; off < SCAN_T; off <<= 1) {
    int t = (tid >= off) ? sm[tid - off] : 0;
    __syncthreads();
    sm[tid] += t;
    __syncthreads();
  }
  return sm[tid] - tsum;   // exclusive prefix
}

__global__ void lgcn_scan_reduce(const int* __restrict__ cnt, int* __restrict__ part, int n) {
  __shared__ int sm[SCAN_T];
  int i0 = blockIdx.x * SCAN_TILE + threadIdx.x * SCAN_I;
  int s = 0;
#pragma unroll
  for (int j = 0; j < SCAN_I; ++j)
    if (i0 + j < n) s += cnt[i0 + j];
  sm[threadIdx.x] = s;
  __syncthreads();
  for (int off = SCAN_T / 2; off > 0; off >>= 1) {
    if (threadIdx.x < off) sm[threadIdx.x] += sm[threadIdx.x + off];
    __syncthreads();
  }
  if (threadIdx.x == 0) part[blockIdx.x] = sm[0];
}

// Single block: exclusive-scan up to 1024 block partials in place.
__global__ void lgcn_scan_partials(int* __restrict__ part, int nb) {
  __shared__ int sm[SCAN_T];
  int i0 = threadIdx.x * SCAN_I;
  int v[SCAN_I];
  int ts = 0;
#pragma unroll
  for (int j = 0; j < SCAN_I; ++j) {
    v[j] = (i0 + j < nb) ? part[i0 + j] : 0;
    ts += v[j];
  }
  int run = block_excl_scan_256(ts, sm);
#pragma unroll
  for (int j = 0; j < SCAN_I; ++j) {
    if (i0 + j < nb) part[i0 + j] = run;
    run += v[j];
  }
}

// Per-block exclusive scan of counts + block offset -> rowptr & cursor.
// cnt_cur is read as counts then overwritten with rowptr (per-thread ownership).
__global__ void lgcn_scan_write(int* __restrict__ cnt_cur, int* __restrict__ rowptr,
                                const int* __restrict__ part, int n, int nnz) {
  __shared__ int sm[SCAN_T];
  int i0 = blockIdx.x * SCAN_TILE + threadIdx.x * SCAN_I;
  int v[SCAN_I];
  int ts = 0;
#pragma unroll
  for (int j = 0; j < SCAN_I; ++j) {
    v[j] = (i0 + j < n) ? cnt_cur[i0 + j] : 0;
    ts += v[j];
  }
  int run = part[blockIdx.x] + block_excl_scan_256(ts, sm);
#pragma unroll
  for (int j = 0; j < SCAN_I; ++j) {
    if (i0 + j < n) { rowptr[i0 + j] = run; cnt_cur[i0 + j] = run; }
    run += v[j];
  }
  if (blockIdx.x == 0 && threadIdx.x == 0) rowptr[n] = nnz;
}

__global__ void lgcn_scatter(const int* __restrict__ src, const int* __restrict__ dst,
                             const float* __restrict__ val, int* __restrict__ cur,
                             long long* __restrict__ pairs, int nnz) {
  int i = blockIdx.x * blockDim.x + threadIdx.x;
  if (i >= nnz) return;
  __builtin_prefetch(dst + i + 8192, 0, 0);          // streaming lookahead
  int d = __builtin_nontemporal_load(dst + i);
  unsigned s = (unsigned)__builtin_nontemporal_load(src + i);
  unsigned vb = __float_as_uint(__builtin_nontemporal_load(val + i));
  int p = atomicAdd(&cur[d], 1);
  pairs[p] = (long long)(((unsigned long long)vb << 32) | (unsigned long long)s);
}

// ----------------------------- atomic fallback ------------------------------
__global__ void lgcn_spmm_atomic(float* out, const int* __restrict__ src,
                                 const int* __restrict__ dst, const float* __restrict__ val,
                                 int nnz, int colPrev, int colCur) {
  int g = blockIdx.x * blockDim.x + threadIdx.x;
  int e = g >> 4, lane = g & 15;
  if (e >= nnz) return;
  int s = src[e];
  int d = dst[e];
  float v = val[e];
  fvec4 gv = *(const fvec4*)(out + colPrev + (size_t)s * OUTC + 4 * lane);
  float* o = out + (size_t)d * OUTC + colCur + 4 * lane;
  atomicAdd(o + 0, v * gv.x);
  atomicAdd(o + 1, v * gv.y);
  atomicAdd(o + 2, v * gv.z);
  atomicAdd(o + 3, v * gv.w);
}

// ----------------------------- launch ---------------------------------------
extern "C" void kernel_launch(void* const* d_in, const int* in_sizes, int n_in,
                              void* d_out, int out_size, void* d_ws, size_t ws_size,
                              hipStream_t stream) {
  const float* user_emb = (const float*)d_in[0];
  const float* item_emb = (const float*)d_in[1];
  const float* edge_val = (const float*)d_in[2];
  const int*   edge_src = (const int*)d_in[3];
  const int*   edge_dst = (const int*)d_in[4];

  const int userN = in_sizes[0] / EMB;
  const int itemN = in_sizes[1] / EMB;
  const int n     = userN + itemN;
  const int nnz   = in_sizes[2];
  float* out = (float*)d_out;

  // Workspace layout (256B-aligned blocks): rowptr[n+1] | cursor[n] | partials | pairs[nnz]
  size_t offRow   = 0;
  size_t offCur   = ((size_t)(n + 1) * 4 + 255) & ~(size_t)255;
  size_t offPart  = (offCur + (size_t)n * 4 + 255) & ~(size_t)255;
  size_t offPairs = (offPart + 4096 + 255) & ~(size_t)255;
  size_t need     = offPairs + (size_t)nnz * 8;
  bool useCsr = (ws_size >= need);

  const int BT = 256;
  int initBlocks = (n * 16 + BT - 1) / BT;
  lgcn_init<<<initBlocks, BT, 0, stream>>>(user_emb, item_emb, out, userN, n);

  if (useCsr) {
    int* rowptr     = (int*)((char*)d_ws + offRow);
    int* cur        = (int*)((char*)d_ws + offCur);
    int* part       = (int*)((char*)d_ws + offPart);
    long long* prs  = (long long*)((char*)d_ws + offPairs);

    int nb = (n + SCAN_TILE - 1) / SCAN_TILE;               // 196 for n=200000
    (void)hipMemsetAsync(cur, 0, (size_t)n * 4, stream);
    lgcn_hist<<<(nnz + BT - 1) / BT, BT, 0, stream>>>(edge_dst, cur, nnz);
    lgcn_scan_reduce<<<nb, SCAN_T, 0, stream>>>(cur, part, n);
    lgcn_scan_partials<<<1, SCAN_T, 0, stream>>>(part, nb);
    lgcn_scan_write<<<nb, SCAN_T, 0, stream>>>(cur, rowptr, part, n, nnz);
    lgcn_scatter<<<(nnz + BT - 1) / BT, BT, 0, stream>>>(edge_src, edge_dst, edge_val,
                                                         cur, prs, nnz);
    int rowBlocks = (n * 16 + BT - 1) / BT;
    for (int k = 1; k <= 3; ++k)
      lgcn_spmm_csr<<<rowBlocks, BT, 0, stream>>>(out, prs, rowptr, n,
                                                  (k - 1) * EMB, k * EMB);
  } else {
    long long tot = (long long)nnz * 16;
    int edgeBlocks = (int)((tot + BT - 1) / BT);
    for (int k = 1; k <= 3; ++k)
      lgcn_spmm_atomic<<<edgeBlocks, BT, 0, stream>>>(out, edge_src, edge_dst, edge_val,
                                                      nnz, (k - 1) * EMB, k * EMB);
  }
}